// GCN_74036646249031
// MI455X (gfx1250) — compile-verified
//
#include <hip/hip_runtime.h>

#define NNODES 20000
#define NFEATD 512
#define NHIDD  1024
#define NEDGE  320000

typedef float v2f __attribute__((ext_vector_type(2)));
typedef float v8f __attribute__((ext_vector_type(8)));

// ---------------------------------------------------------------------------
// C[M,N] = A[M,K] @ B[K,N], row-major, fp32, via V_WMMA_F32_16X16X4_F32.
// One wave owns a 32x64 tile: 2 M-subtiles x 4 N-subtiles, K stepped by 4.
// ---------------------------------------------------------------------------
template <int K, int NOUT>
__global__ __launch_bounds__(256) void gemm_wmma_f32(const float* __restrict__ A,
                                                     const float* __restrict__ B,
                                                     float* __restrict__ C) {
  const int lane      = threadIdx.x & 31;
  const int waveInBlk = threadIdx.x >> 5;
  const int waveId    = blockIdx.x * 8 + waveInBlk;

  const int ntilesN = NOUT / 64;
  const int mstrip  = waveId / ntilesN;
  const int nstrip  = waveId % ntilesN;
  const int m0 = mstrip * 32;
  const int n0 = nstrip * 64;
  if (m0 >= NNODES) return;

  const int half = lane >> 4;   // 0 -> K,K+1 ; 1 -> K+2,K+3
  const int r    = lane & 15;

  v8f acc[2][4];
#pragma unroll
  for (int mi = 0; mi < 2; ++mi)
#pragma unroll
    for (int ni = 0; ni < 4; ++ni)
      acc[mi][ni] = (v8f){0.f, 0.f, 0.f, 0.f, 0.f, 0.f, 0.f, 0.f};

  for (int k = 0; k < K; k += 4) {
    v2f afrag[2];
#pragma unroll
    for (int mi = 0; mi < 2; ++mi) {
      const float* ap = A + (size_t)(m0 + mi * 16 + r) * K + k + 2 * half;
      afrag[mi][0] = ap[0];
      afrag[mi][1] = ap[1];
    }
    v2f bfrag[4];
#pragma unroll
    for (int ni = 0; ni < 4; ++ni) {
      const float* bp = B + (size_t)(k + 2 * half) * NOUT + n0 + ni * 16 + r;
      bfrag[ni][0] = bp[0];
      bfrag[ni][1] = bp[NOUT];
    }
#pragma unroll
    for (int mi = 0; mi < 2; ++mi)
#pragma unroll
      for (int ni = 0; ni < 4; ++ni)
        acc[mi][ni] = __builtin_amdgcn_wmma_f32_16x16x4_f32(
            false, afrag[mi], false, bfrag[ni], (short)0, acc[mi][ni], false, false);
  }

#pragma unroll
  for (int mi = 0; mi < 2; ++mi)
#pragma unroll
    for (int ni = 0; ni < 4; ++ni)
#pragma unroll
      for (int v = 0; v < 8; ++v)
        C[(size_t)(m0 + mi * 16 + v + 8 * half) * NOUT + n0 + ni * 16 + r] =
            acc[mi][ni][v];
}

// ---------------------------------------------------------------------------
// Zero a float buffer (float4 granularity; count4 = floats/4).
// ---------------------------------------------------------------------------
__global__ __launch_bounds__(256) void zero_f4(float4* __restrict__ p, int count4) {
  int i = blockIdx.x * 256 + threadIdx.x;
  if (i < count4) p[i] = (float4){0.f, 0.f, 0.f, 0.f};
}

// ---------------------------------------------------------------------------
// One block per edge: agg[dst[e], :] += adj_vals[e] * sup[src[e], :]
// float4 gather reads (L2-resident), per-float global_atomic_add_f32.
// ---------------------------------------------------------------------------
__global__ __launch_bounds__(256) void scatter_edges(const float* __restrict__ sup,
                                                     const float* __restrict__ vals,
                                                     const int* __restrict__ src,
                                                     const int* __restrict__ dst,
                                                     float* __restrict__ agg,
                                                     int feat) {
  const int e = blockIdx.x;
  const float v = vals[e];
  const size_t srow = (size_t)src[e] * feat;
  const size_t drow = (size_t)dst[e] * feat;
  const float4* sp = (const float4*)(sup + srow);
  float* ap = agg + drow;
  const int nvec = feat >> 2;
  for (int i = threadIdx.x; i < nvec; i += 256) {
    float4 t = sp[i];
    atomicAdd(ap + 4 * i + 0, v * t.x);
    atomicAdd(ap + 4 * i + 1, v * t.y);
    atomicAdd(ap + 4 * i + 2, v * t.z);
    atomicAdd(ap + 4 * i + 3, v * t.w);
  }
}

// ---------------------------------------------------------------------------
// h = relu(agg1 + b1) * inverted-dropout(u_drop), in place on agg1.
// ---------------------------------------------------------------------------
__global__ __launch_bounds__(256) void finish_hidden(float* __restrict__ h,
                                                     const float* __restrict__ b1,
                                                     const float* __restrict__ u) {
  size_t i = (size_t)blockIdx.x * 256 + threadIdx.x;
  if (i >= (size_t)NNODES * NHIDD) return;
  int j = (int)(i & (NHIDD - 1));
  float val = h[i] + b1[j];
  val = val > 0.f ? val : 0.f;
  float keep = (u[i] > 0.5f) ? 2.0f : 0.0f;  // 1/(1-p) with p=0.5
  h[i] = val * keep;
}

// ---------------------------------------------------------------------------
// out[i, j] = b2[j]  (bias pre-seeded; second scatter accumulates on top).
// ---------------------------------------------------------------------------
__global__ __launch_bounds__(256) void init_out_bias(float* __restrict__ out,
                                                     const float* __restrict__ b2) {
  size_t i = (size_t)blockIdx.x * 256 + threadIdx.x;
  if (i >= (size_t)NNODES * NFEATD) return;
  out[i] = b2[i & (NFEATD - 1)];
}

extern "C" void kernel_launch(void* const* d_in, const int* in_sizes, int n_in,
                              void* d_out, int out_size, void* d_ws, size_t ws_size,
                              hipStream_t stream) {
  (void)in_sizes; (void)n_in; (void)out_size; (void)ws_size;

  const float* x    = (const float*)d_in[0];
  const float* W1   = (const float*)d_in[1];
  const float* b1   = (const float*)d_in[2];
  const float* W2   = (const float*)d_in[3];
  const float* b2   = (const float*)d_in[4];
  const float* adjv = (const float*)d_in[5];
  const float* udrp = (const float*)d_in[6];
  const int*   src  = (const int*)d_in[7];
  const int*   dst  = (const int*)d_in[8];
  float*       out  = (float*)d_out;

  float* sup = (float*)d_ws;                      // N*NHID floats (reused as support2)
  float* agg = sup + (size_t)NNODES * NHIDD;      // N*NHID floats (agg1, then h)

  // 1) support1 = x @ W1   [20000x512] x [512x1024]
  {
    const int waves = (NNODES / 32) * (NHIDD / 64);     // 10000
    gemm_wmma_f32<NFEATD, NHIDD><<<waves / 8, 256, 0, stream>>>(x, W1, sup);
  }
  // 2) agg1 = 0
  {
    const int count4 = NNODES * NHIDD / 4;              // 5,120,000
    zero_f4<<<(count4 + 255) / 256, 256, 0, stream>>>((float4*)agg, count4);
  }
  // 3) agg1[dst] += adj * support1[src]
  scatter_edges<<<NEDGE, 256, 0, stream>>>(sup, adjv, src, dst, agg, NHIDD);
  // 4) h = relu(agg1 + b1) * dropout_mask   (in place)
  {
    const size_t total = (size_t)NNODES * NHIDD;
    finish_hidden<<<(unsigned)((total + 255) / 256), 256, 0, stream>>>(agg, b1, udrp);
  }
  // 5) support2 = h @ W2   [20000x1024] x [1024x512]  (into sup buffer)
  {
    const int waves = (NNODES / 32) * (NFEATD / 64);    // 5000
    gemm_wmma_f32<NHIDD, NFEATD><<<waves / 8, 256, 0, stream>>>(agg, W2, sup);
  }
  // 6) out = broadcast(b2)
  {
    const size_t total = (size_t)NNODES * NFEATD;
    init_out_bias<<<(unsigned)((total + 255) / 256), 256, 0, stream>>>(out, b2);
  }
  // 7) out[dst] += adj * support2[src]
  scatter_edges<<<NEDGE, 256, 0, stream>>>(sup, adjv, src, dst, out, NFEATD);
}